// SEC_87574383165526
// MI455X (gfx1250) — compile-verified
//
#include <hip/hip_runtime.h>
#include <math.h>

typedef __attribute__((ext_vector_type(2))) float v2f;
typedef __attribute__((ext_vector_type(8))) float v8f;

#define NCOLS 4096
#define COLS_PER_ITER 128            // 32 lanes * 4 floats
#define NITERS (NCOLS / COLS_PER_ITER)  // 32
#define WAVES_PER_BLOCK 8
#define STAGE_FLOATS 128             // one stage buffer per wave
#define STAGE_BYTES (STAGE_FLOATS * 4)

// One wave per row. Async global->LDS double-buffered streaming of scores,
// labels via regular coalesced b128 loads, row sums accumulated with
// v_wmma_f32_16x16x4_f32 (B = ones), masked max on VALU.
__global__ __launch_bounds__(256) void sec_row_kernel(
    const float* __restrict__ scores,
    const int*   __restrict__ label,
    float* __restrict__ row_loss)
{
    __shared__ float sbuf[WAVES_PER_BLOCK * 2 * STAGE_FLOATS];

    const int lane = threadIdx.x & 31;
    const int wave = threadIdx.x >> 5;
    const int row  = blockIdx.x * WAVES_PER_BLOCK + wave;

    const unsigned ldsWaveBase =
        (unsigned)(uintptr_t)(void*)(&sbuf[wave * 2 * STAGE_FLOATS]);
    const unsigned long long rowPtr =
        (unsigned long long)(const void*)(scores + (size_t)row * NCOLS);
    const int4* __restrict__ lrow =
        (const int4*)(label + (size_t)row * NCOLS);

    // Prologue: async-load stage 0 into buffer 0 (16B per lane, coalesced).
    {
        unsigned lds0 = ldsWaveBase + (unsigned)lane * 16u;
        unsigned g0   = (unsigned)lane * 16u;
        asm volatile("global_load_async_to_lds_b128 %0, %1, %2"
                     :: "v"(lds0), "v"(g0), "s"(rowPtr) : "memory");
    }

    v8f cden = {0.f, 0.f, 0.f, 0.f, 0.f, 0.f, 0.f, 0.f};
    v8f cpos = {0.f, 0.f, 0.f, 0.f, 0.f, 0.f, 0.f, 0.f};
    const v2f bones = {1.0f, 1.0f};
    float mneg = -INFINITY;

    for (int it = 0; it < NITERS; ++it) {
        const int buf = it & 1;

        // Prefetch next stage into the other buffer, then wait so the
        // current buffer's data is resident (async loads complete in order).
        if (it + 1 < NITERS) {
            unsigned lds = ldsWaveBase + (unsigned)((it + 1) & 1) * STAGE_BYTES
                         + (unsigned)lane * 16u;
            unsigned g = (unsigned)((it + 1) * COLS_PER_ITER + lane * 4) * 4u;
            asm volatile("global_load_async_to_lds_b128 %0, %1, %2"
                         :: "v"(lds), "v"(g), "s"(rowPtr) : "memory");
            asm volatile("s_wait_asynccnt 1" ::: "memory");
        } else {
            asm volatile("s_wait_asynccnt 0" ::: "memory");
        }

        const float4 s4 =
            *(const float4*)&sbuf[wave * 2 * STAGE_FLOATS + buf * STAGE_FLOATS
                                  + lane * 4];
        const int4 lb = lrow[it * 32 + lane];

        const float ex = __expf(s4.x);
        const float ey = __expf(s4.y);
        const float ez = __expf(s4.z);
        const float ew = __expf(s4.w);

        const float px = (lb.x > 0) ? ex : 0.0f;
        const float py = (lb.y > 0) ? ey : 0.0f;
        const float pz = (lb.z > 0) ? ez : 0.0f;
        const float pw = (lb.w > 0) ? ew : 0.0f;

        const float nx = (lb.x > 0) ? -INFINITY : s4.x;
        const float ny = (lb.y > 0) ? -INFINITY : s4.y;
        const float nz = (lb.z > 0) ? -INFINITY : s4.z;
        const float nw = (lb.w > 0) ? -INFINITY : s4.w;
        mneg = fmaxf(mneg, fmaxf(fmaxf(nx, ny), fmaxf(nz, nw)));

        // A-matrix 16x4 f32 layout: lane m holds (K0,K1), lane m+16 (K2,K3).
        // With B == ones, D(m,n) += sum_k A(m,k): row-sum on the matrix pipe.
        v2f a0 = {ex, ey};
        v2f a1 = {ez, ew};
        cden = __builtin_amdgcn_wmma_f32_16x16x4_f32(
            false, a0, false, bones, (short)0, cden, false, false);
        cden = __builtin_amdgcn_wmma_f32_16x16x4_f32(
            false, a1, false, bones, (short)0, cden, false, false);

        v2f p0 = {px, py};
        v2f p1 = {pz, pw};
        cpos = __builtin_amdgcn_wmma_f32_16x16x4_f32(
            false, p0, false, bones, (short)0, cpos, false, false);
        cpos = __builtin_amdgcn_wmma_f32_16x16x4_f32(
            false, p1, false, bones, (short)0, cpos, false, false);
    }

    // C/D 16x16 layout: lane<16 vgpr r = D(m=r, n=lane); lane>=16 m=r+8.
    // Sum of a lane's 8 accumulators = half-sum; xor-16 shuffle completes it.
    float sden = cden[0] + cden[1] + cden[2] + cden[3]
               + cden[4] + cden[5] + cden[6] + cden[7];
    sden += __shfl_xor(sden, 16, 32);

    float spos = cpos[0] + cpos[1] + cpos[2] + cpos[3]
               + cpos[4] + cpos[5] + cpos[6] + cpos[7];
    spos += __shfl_xor(spos, 16, 32);

    for (int off = 16; off > 0; off >>= 1)
        mneg = fmaxf(mneg, __shfl_xor(mneg, off, 32));

    if (lane == 0) {
        const float pos = spos + __expf(mneg);  // exp(max) == max(exp), exp monotone
        row_loss[row] = -__logf(pos / sden + 0.05f);
    }
}

__global__ __launch_bounds__(256) void sec_mean_kernel(
    const float* __restrict__ rl, float* __restrict__ out, int n)
{
    __shared__ float red[256];
    float s = 0.0f;
    for (int i = threadIdx.x; i < n; i += 256) s += rl[i];
    red[threadIdx.x] = s;
    __syncthreads();
    for (int w = 128; w > 0; w >>= 1) {
        if ((int)threadIdx.x < w) red[threadIdx.x] += red[threadIdx.x + w];
        __syncthreads();
    }
    if (threadIdx.x == 0) out[0] = red[0] / (float)n;
}

extern "C" void kernel_launch(void* const* d_in, const int* in_sizes, int n_in,
                              void* d_out, int out_size, void* d_ws, size_t ws_size,
                              hipStream_t stream) {
    const float* scores = (const float*)d_in[0];
    const int*   label  = (const int*)d_in[1];
    float* row_loss = (float*)d_ws;

    const int N = in_sizes[0] / NCOLS;  // 8192 rows

    sec_row_kernel<<<N / WAVES_PER_BLOCK, 256, 0, stream>>>(scores, label, row_loss);
    sec_mean_kernel<<<1, 256, 0, stream>>>(row_loss, (float*)d_out, N);
}